// STViG_83296595739386
// MI455X (gfx1250) — compile-verified
//
#include <hip/hip_runtime.h>
#include <math.h>

typedef __attribute__((ext_vector_type(2))) float v2f;
typedef __attribute__((ext_vector_type(8))) float v8f;

#define INVC 0.999995000037499726f   // 1/sqrt(1+1e-5)

// ---------------------------------------------------------------------------
// fp32 WMMA: D(16x16) += A(16x4) * B(4x16), all f32.
// A frag: lane<16 -> {A[m][k0],A[m][k1]}, lane>=16 -> {A[m][k2],A[m][k3]}, m=lane&15
// B frag: lane<16 -> {B[k0][n],B[k1][n]}, lane>=16 -> {B[k2][n],B[k3][n]}, n=lane&15
// With W stored row-major (N x K), B[k][n] = W[n][k] -> contiguous b64 loads.
// ---------------------------------------------------------------------------
__device__ __forceinline__ v8f wmma_f32_k4(v2f a, v2f b, v8f c) {
  return __builtin_amdgcn_wmma_f32_16x16x4_f32(false, a, false, b, (short)0, c,
                                               false, false);
}

// ---------------------------------------------------------------------------
// Generic NT GEMM: Y[m,n] = epi( (sum_k preA(A[rowA(m),k]) * W[n,k] + bias[n]) )
//   preA (PRE):  a -> relu(a + cst[k])           (fused grapher)
//   rowA map:    m=(b,tq,p) over (Tp,Pdim); rowA=(b*T + tq*st)*Pdim + p
//                identity when Pdim=Tp=st=T=1
//   epi:         v=(acc+bias)*INV; if ADD v+=addend[m,n]; if RELU v=max(v,0)
// One wave computes a 32(M) x 64(N) strip: 2x4 accumulator tiles
// (2 A-frags amortize the 4 B-frag loads -> 0.75 b64 loads per WMMA).
// Requires: M%32==0, N%64==0, K%4==0.
// ---------------------------------------------------------------------------
template <bool PRE, bool ADD, bool RELU>
__global__ __launch_bounds__(256) void stvig_gemm_wmma(
    const float* __restrict__ A, const float* __restrict__ W,
    const float* __restrict__ bias, const float* __restrict__ addend,
    const float* __restrict__ cst, float* __restrict__ Y,
    int M, int N, int K, int Pdim, int Tp, int st, int T) {
  const int wave = blockIdx.x * (blockDim.x >> 5) + (threadIdx.x >> 5);
  const int lane = threadIdx.x & 31;
  const int half = lane >> 4;
  const int lr   = lane & 15;
  const int tilesM = M >> 5;
  const int tilesN = N >> 6;
  if (wave >= tilesM * tilesN) return;
  const int tm = (wave % tilesM) << 5;
  const int tn = (wave / tilesM) << 6;

  // per-lane A rows (optional strided-T gather for the downsample path)
  long rowA[2];
#pragma unroll
  for (int u = 0; u < 2; ++u) {
    const int m  = tm + 16 * u + lr;
    const int bI = m / (Tp * Pdim);
    const int rm = m - bI * (Tp * Pdim);
    const int tq = rm / Pdim;
    const int pp = rm - tq * Pdim;
    rowA[u] = (long)(bI * T + tq * st) * Pdim + pp;
  }

  const float* Ap0 = A + rowA[0] * (long)K + 2 * half;
  const float* Ap1 = A + rowA[1] * (long)K + 2 * half;
  const float* Wp0 = W + (long)(tn + lr) * K + 2 * half;
  const float* Wp1 = Wp0 + 16L * K;
  const float* Wp2 = Wp1 + 16L * K;
  const float* Wp3 = Wp2 + 16L * K;
  const float* Cp  = PRE ? (cst + 2 * half) : nullptr;

  v8f acc[2][4] = {};
#pragma unroll 2
  for (int k = 0; k < K; k += 4) {
    v2f a0 = *(const v2f*)Ap0;  Ap0 += 4;
    v2f a1 = *(const v2f*)Ap1;  Ap1 += 4;
    if (PRE) {
      v2f cv = *(const v2f*)Cp;  Cp += 4;
      a0.x = fmaxf(a0.x + cv.x, 0.0f);
      a0.y = fmaxf(a0.y + cv.y, 0.0f);
      a1.x = fmaxf(a1.x + cv.x, 0.0f);
      a1.y = fmaxf(a1.y + cv.y, 0.0f);
    }
    v2f b0 = *(const v2f*)Wp0;  Wp0 += 4;
    v2f b1 = *(const v2f*)Wp1;  Wp1 += 4;
    v2f b2 = *(const v2f*)Wp2;  Wp2 += 4;
    v2f b3 = *(const v2f*)Wp3;  Wp3 += 4;
    acc[0][0] = wmma_f32_k4(a0, b0, acc[0][0]);
    acc[1][0] = wmma_f32_k4(a1, b0, acc[1][0]);
    acc[0][1] = wmma_f32_k4(a0, b1, acc[0][1]);
    acc[1][1] = wmma_f32_k4(a1, b1, acc[1][1]);
    acc[0][2] = wmma_f32_k4(a0, b2, acc[0][2]);
    acc[1][2] = wmma_f32_k4(a1, b2, acc[1][2]);
    acc[0][3] = wmma_f32_k4(a0, b3, acc[0][3]);
    acc[1][3] = wmma_f32_k4(a1, b3, acc[1][3]);
  }

#pragma unroll
  for (int u = 0; u < 2; ++u) {
#pragma unroll
    for (int t = 0; t < 4; ++t) {
      const int n = tn + 16 * t + lr;
      const float bz = bias[n];
#pragma unroll
      for (int v = 0; v < 8; ++v) {
        const int mr = tm + 16 * u + v + 8 * half;  // C/D layout: VGPR v -> M row
        float val = (acc[u][t][v] + bz) * INVC;
        if (ADD)  val += addend[(long)mr * N + n];
        if (RELU) val = fmaxf(val, 0.0f);
        Y[(long)mr * N + n] = val;
      }
    }
  }
}

// ---------------------------------------------------------------------------
// 3x3 conv (edge padding in T,P; stride (st,1)) as implicit GEMM, K = 9*C.
// H is channel-last (B,T,Pp,C); W2r is repacked (N x 9*C) with k = j*C + c.
// Y[m,n] = relu((acc + bias[n]) * INV), m over (B,Tp,Pp), N = C.
// 32(M) x 64(N) per wave, same as the GEMM.
// ---------------------------------------------------------------------------
__global__ __launch_bounds__(256) void stvig_conv3x3_wmma(
    const float* __restrict__ H, const float* __restrict__ W2r,
    const float* __restrict__ bias, float* __restrict__ Y,
    int B, int T, int Tp, int Pp, int C, int st) {
  const int N = C;
  const int K = 9 * C;
  const int M = B * Tp * Pp;
  const int wave = blockIdx.x * (blockDim.x >> 5) + (threadIdx.x >> 5);
  const int lane = threadIdx.x & 31;
  const int half = lane >> 4;
  const int lr   = lane & 15;
  const int tilesM = M >> 5;
  const int tilesN = N >> 6;
  if (wave >= tilesM * tilesN) return;
  const int tm = (wave % tilesM) << 5;
  const int tn = (wave / tilesM) << 6;

  int bI[2], tq[2], pp[2];
#pragma unroll
  for (int u = 0; u < 2; ++u) {
    const int m = tm + 16 * u + lr;
    bI[u] = m / (Tp * Pp);
    const int rm = m - bI[u] * (Tp * Pp);
    tq[u] = rm / Pp;
    pp[u] = rm - tq[u] * Pp;
  }

  const long wrow = (long)(tn + lr) * K + 2 * half;
  v8f acc[2][4] = {};

#pragma unroll
  for (int j = 0; j < 9; ++j) {
    const int dy = j / 3 - 1;
    const int dx = j % 3 - 1;
    const float* Ap[2];
#pragma unroll
    for (int u = 0; u < 2; ++u) {
      int ts = tq[u] * st + dy;  ts = ts < 0 ? 0 : (ts > T - 1 ? T - 1 : ts);
      int ps = pp[u] + dx;       ps = ps < 0 ? 0 : (ps > Pp - 1 ? Pp - 1 : ps);
      Ap[u] = H + ((long)(bI[u] * T + ts) * Pp + ps) * C + 2 * half;
    }
    const float* Wj0 = W2r + wrow + j * C;
    const float* Wj1 = Wj0 + 16L * K;
    const float* Wj2 = Wj1 + 16L * K;
    const float* Wj3 = Wj2 + 16L * K;
#pragma unroll 2
    for (int c = 0; c < C; c += 4) {
      v2f a0 = *(const v2f*)(Ap[0] + c);
      v2f a1 = *(const v2f*)(Ap[1] + c);
      v2f b0 = *(const v2f*)(Wj0 + c);
      v2f b1 = *(const v2f*)(Wj1 + c);
      v2f b2 = *(const v2f*)(Wj2 + c);
      v2f b3 = *(const v2f*)(Wj3 + c);
      acc[0][0] = wmma_f32_k4(a0, b0, acc[0][0]);
      acc[1][0] = wmma_f32_k4(a1, b0, acc[1][0]);
      acc[0][1] = wmma_f32_k4(a0, b1, acc[0][1]);
      acc[1][1] = wmma_f32_k4(a1, b1, acc[1][1]);
      acc[0][2] = wmma_f32_k4(a0, b2, acc[0][2]);
      acc[1][2] = wmma_f32_k4(a1, b2, acc[1][2]);
      acc[0][3] = wmma_f32_k4(a0, b3, acc[0][3]);
      acc[1][3] = wmma_f32_k4(a1, b3, acc[1][3]);
    }
  }

#pragma unroll
  for (int u = 0; u < 2; ++u) {
#pragma unroll
    for (int t = 0; t < 4; ++t) {
      const int n = tn + 16 * t + lr;
      const float bz = bias[n];
#pragma unroll
      for (int v = 0; v < 8; ++v) {
        const int mr = tm + 16 * u + v + 8 * half;
        float val = fmaxf((acc[u][t][v] + bz) * INVC, 0.0f);
        Y[(long)mr * N + n] = val;
      }
    }
  }
}

// --- small helper kernels ---------------------------------------------------

// positional encoding: PE[n,o] = (kpts[n,:] . pe_w[o,:] + pe_b[o]) * INV
__global__ void stvig_pe_kernel(const float* __restrict__ kpts,
                                const float* __restrict__ pe_w,
                                const float* __restrict__ pe_b,
                                float* __restrict__ PE, int NP) {
  int i = blockIdx.x * blockDim.x + threadIdx.x;
  if (i >= NP * 64) return;
  int n = i >> 6, o = i & 63;
  const float* kp = kpts + 3 * n;
  float v = kp[0] * pe_w[o * 3] + kp[1] * pe_w[o * 3 + 1] +
            kp[2] * pe_w[o * 3 + 2] + pe_b[o];
  PE[i] = v * INVC;
}

// grapher collapses to x -> relu(x + cst):
// cst[o] = (sum_j fc4_w[o,j] * relu(intra_b[j]*INV) + fc4_b[o]) * INV
__global__ void stvig_cst_kernel(const float* __restrict__ fc4_w,
                                 const float* __restrict__ fc4_b,
                                 const float* __restrict__ intra_b,
                                 float* __restrict__ cst, int cin, int twoOut) {
  int o = blockIdx.x * blockDim.x + threadIdx.x;
  if (o >= cin) return;
  float s = fc4_b[o];
  for (int j = 0; j < twoOut; ++j)
    s += fc4_w[o * twoOut + j] * fmaxf(intra_b[j] * INVC, 0.0f);
  cst[o] = s * INVC;
}

// repack conv weight (O,C,3,3) -> (O, 9*C) with k = j*C + c
__global__ void stvig_repack_kernel(const float* __restrict__ w,
                                    float* __restrict__ wr, int O, int C) {
  int i = blockIdx.x * blockDim.x + threadIdx.x;
  if (i >= O * C * 9) return;
  int o = i / (C * 9);
  int r = i - o * (C * 9);
  int j = r / C;
  int c = r - j * C;
  wr[(long)o * C * 9 + j * C + c] = w[(long)o * C * 9 + c * 9 + j];
}

// mean over (T=15, P=15): pooled[b,c] = mean_tp X[b,t,p,c]
__global__ void stvig_pool_kernel(const float* __restrict__ X,
                                  float* __restrict__ pooled) {
  int i = blockIdx.x * blockDim.x + threadIdx.x;
  if (i >= 32 * 512) return;
  int b = i >> 9, c = i & 511;
  const float* base = X + (long)b * 225 * 512 + c;
  float s = 0.0f;
  for (int r = 0; r < 225; ++r) s += base[r * 512];
  pooled[i] = s * (1.0f / 225.0f);
}

// logit + sigmoid
__global__ void stvig_head_kernel(const float* __restrict__ h,
                                  const float* __restrict__ h2_w,
                                  const float* __restrict__ h2_b,
                                  float* __restrict__ out) {
  int b = threadIdx.x;
  if (b >= 32) return;
  float s = h2_b[0];
  for (int c = 0; c < 256; ++c) s += h[b * 256 + c] * h2_w[c];
  out[b] = 1.0f / (1.0f + expf(-s));
}

// ---------------------------------------------------------------------------
extern "C" void kernel_launch(void* const* d_in, const int* in_sizes, int n_in,
                              void* d_out, int out_size, void* d_ws,
                              size_t ws_size, hipStream_t stream) {
  (void)in_sizes; (void)n_in; (void)out_size; (void)ws_size;
  // JAX pytree leaf order (dict keys sorted alphabetically, lists in order):
  // 0 inputs, 1 kpts,
  // graphers[g] @ 2+12g: fc1_b,fc1_w,fc2_b,fc2_w,fc3_b,fc3_w,fc4_b,fc4_w,
  //                      inter_b,inter_w,intra_b,intra_w
  // 38 h1_b, 39 h1_w, 40 h2_b, 41 h2_w
  // parts[s] @ 42+8s: c1_b,c1_w,c2_b,c2_w,c3_b,c3_w,ds_b,ds_w
  // 66 pe_b, 67 pe_w, 68 stem_b, 69 stem_w
  auto prm = [&](int i) { return (const float*)d_in[i]; };
  const float* inputs = (const float*)d_in[0];
  const float* kpts   = (const float*)d_in[1];

  float* ws = (float*)d_ws;
  float* PE = ws;                 // 14400*64
  float* XA = PE + 921600;        // up to 7200*512
  float* XB = XA + 3686400;
  float* H1 = XB + 3686400;       // up to 14400*128
  float* H2 = H1 + 1843200;       // up to 14400*64 / 7200*128
  float* H3 = H2 + 921600;        // up to 14400*256 / 7200*512
  float* W2 = H3 + 3686400;       // up to 128*128*9
  float* C1 = W2 + 147456;        // 64
  float* C2 = C1 + 64;            // 256
  float* C3 = C2 + 256;           // 512
  float* PL = C3 + 512;           // 32*512
  float* HH = PL + 16384;         // 32*256

  const dim3 blk(256);
  auto nblk = [](int M, int N) {
    int tiles = (M >> 5) * (N >> 6);
    return dim3((unsigned)((tiles + 7) / 8));
  };

  // --- precomputations ---
  stvig_pe_kernel<<<dim3(3600), blk, 0, stream>>>(kpts, prm(67), prm(66), PE, 14400);
  stvig_cst_kernel<<<dim3(1), dim3(64),  0, stream>>>(prm(2 + 7),  prm(2 + 6),  prm(2 + 10),  C1,  64, 128);
  stvig_cst_kernel<<<dim3(1), dim3(256), 0, stream>>>(prm(14 + 7), prm(14 + 6), prm(14 + 10), C2, 256, 256);
  stvig_cst_kernel<<<dim3(2), dim3(256), 0, stream>>>(prm(26 + 7), prm(26 + 6), prm(26 + 10), C3, 512, 256);

  // --- stem: X0 = (inputs @ stem_w^T + stem_b)*INV + PE  (14400x3072 @ 3072x64)
  stvig_gemm_wmma<false, true, false><<<nblk(14400, 64), blk, 0, stream>>>(
      inputs, prm(69), prm(68), PE, nullptr, XA, 14400, 64, 3072, 1, 1, 1, 1);

  // --- stage 1: cin=64, out=64, st=1, T=30 -> Tp=30 ---
  stvig_repack_kernel<<<dim3(144), blk, 0, stream>>>(prm(42 + 3), W2, 64, 64);
  stvig_gemm_wmma<true, false, true><<<nblk(14400, 64), blk, 0, stream>>>(
      XA, prm(42 + 1), prm(42 + 0), nullptr, C1, H1, 14400, 64, 64, 1, 1, 1, 1);
  stvig_conv3x3_wmma<<<nblk(14400, 64), blk, 0, stream>>>(H1, W2, prm(42 + 2), H2,
                                                          32, 30, 30, 15, 64, 1);
  stvig_gemm_wmma<false, false, true><<<nblk(14400, 256), blk, 0, stream>>>(
      H2, prm(42 + 5), prm(42 + 4), nullptr, nullptr, H3, 14400, 256, 64, 1, 1, 1, 1);
  stvig_gemm_wmma<true, true, true><<<nblk(14400, 256), blk, 0, stream>>>(
      XA, prm(42 + 7), prm(42 + 6), H3, C1, XB, 14400, 256, 64, 1, 1, 1, 1);

  // --- stage 2: cin=256, out=128, st=2, T=30 -> Tp=15 ---
  stvig_repack_kernel<<<dim3(576), blk, 0, stream>>>(prm(50 + 3), W2, 128, 128);
  stvig_gemm_wmma<true, false, true><<<nblk(14400, 128), blk, 0, stream>>>(
      XB, prm(50 + 1), prm(50 + 0), nullptr, C2, H1, 14400, 128, 256, 1, 1, 1, 1);
  stvig_conv3x3_wmma<<<nblk(7200, 128), blk, 0, stream>>>(H1, W2, prm(50 + 2), H2,
                                                          32, 30, 15, 15, 128, 2);
  stvig_gemm_wmma<false, false, true><<<nblk(7200, 512), blk, 0, stream>>>(
      H2, prm(50 + 5), prm(50 + 4), nullptr, nullptr, H3, 7200, 512, 128, 1, 1, 1, 1);
  stvig_gemm_wmma<true, true, true><<<nblk(7200, 512), blk, 0, stream>>>(
      XB, prm(50 + 7), prm(50 + 6), H3, C2, XA, 7200, 512, 256, 15, 15, 2, 30);

  // --- stage 3: cin=512, out=128, st=1, T=15 -> Tp=15 ---
  stvig_repack_kernel<<<dim3(576), blk, 0, stream>>>(prm(58 + 3), W2, 128, 128);
  stvig_gemm_wmma<true, false, true><<<nblk(7200, 128), blk, 0, stream>>>(
      XA, prm(58 + 1), prm(58 + 0), nullptr, C3, H1, 7200, 128, 512, 1, 1, 1, 1);
  stvig_conv3x3_wmma<<<nblk(7200, 128), blk, 0, stream>>>(H1, W2, prm(58 + 2), H2,
                                                          32, 15, 15, 15, 128, 1);
  stvig_gemm_wmma<false, false, true><<<nblk(7200, 512), blk, 0, stream>>>(
      H2, prm(58 + 5), prm(58 + 4), nullptr, nullptr, H3, 7200, 512, 128, 1, 1, 1, 1);
  stvig_gemm_wmma<true, true, true><<<nblk(7200, 512), blk, 0, stream>>>(
      XA, prm(58 + 7), prm(58 + 6), H3, C3, XB, 7200, 512, 512, 1, 1, 1, 1);

  // --- head ---
  stvig_pool_kernel<<<dim3(64), blk, 0, stream>>>(XB, PL);
  stvig_gemm_wmma<false, false, true><<<dim3(1), blk, 0, stream>>>(
      PL, prm(39), prm(38), nullptr, nullptr, HH, 32, 256, 512, 1, 1, 1, 1);
  stvig_head_kernel<<<dim3(1), dim3(32), 0, stream>>>(HH, prm(41), prm(40),
                                                      (float*)d_out);
}